// Quantizer_10033043603468
// MI455X (gfx1250) — compile-verified
//
#include <hip/hip_runtime.h>

typedef __attribute__((ext_vector_type(2))) float v2f;
typedef __attribute__((ext_vector_type(8))) float v8f;

#define NVEC   65536      // B*H*W
#define KCODES 1024
#define DDIM   64
#define NELEM  4194304    // B*L*H*W
#define PITCH  68         // LDS row pitch (floats): conflict-free b64 fragment reads

// ---------------------------------------------------------------------------
// Kernel A: fp32 WMMA distance matmul + per-row argmin.
// One wave handles 16 vectors vs all 1024 codes (64 tiles of 16 codes).
// d(n,c) = ||e_c||^2 - 2 z_n.e_c   (||z||^2 constant per row, dropped)
// ---------------------------------------------------------------------------
__global__ __launch_bounds__(256) void vq_argmin_kernel(
    const float* __restrict__ ze, const float* __restrict__ emb,
    int* __restrict__ wsIdx, float* __restrict__ outIdxF)
{
    extern __shared__ float smem[];
    float* semb  = smem;                       // KCODES * PITCH floats
    float* snorm = smem + KCODES * PITCH;      // KCODES floats

    const int tid = threadIdx.x;

    // Stage codebook into LDS (coalesced global reads, pitched rows)
    for (int i = tid; i < KCODES * DDIM; i += 256)
        semb[(i >> 6) * PITCH + (i & 63)] = emb[i];
    __syncthreads();

    // Per-code squared norms
    for (int c = tid; c < KCODES; c += 256) {
        const float* r = &semb[c * PITCH];
        float s = 0.f;
        #pragma unroll
        for (int k = 0; k < DDIM; ++k) s += r[k] * r[k];
        snorm[c] = s;
    }
    __syncthreads();

    const int lane = tid & 31;
    const int wave = tid >> 5;
    const int wid  = blockIdx.x * 8 + wave;
    const int n0   = wid << 4;          // first of 16 vectors for this wave
    const int col  = lane & 15;         // A row m / B col n / C col n
    const int kg   = lane >> 4;         // K-pair group (ISA fp32 A/B layout)

    // ze element (n,l) lives at (n>>12)*262144 + (n&4095) + l*4096
    const float* zb = ze + (size_t)(n0 >> 12) * 262144u + (size_t)(n0 & 4095) + col;

    // Load all 16 A fragments (16x4 fp32 each = 2 VGPRs) once, reuse 64x
    v2f a[16];
    #pragma unroll
    for (int t = 0; t < 16; ++t) {
        const int k0 = 4 * t + 2 * kg;
        a[t][0] = zb[(size_t)k0 * 4096u];
        a[t][1] = zb[(size_t)(k0 + 1) * 4096u];
    }

    float best[8];
    int   bidx[8];
    #pragma unroll
    for (int i = 0; i < 8; ++i) { best[i] = 3.4e38f; bidx[i] = 0; }

    for (int j = 0; j < 64; ++j) {
        const int c0 = j << 4;
        const float* bp = &semb[(size_t)(c0 + col) * PITCH + 2 * kg];
        v8f acc = {};
        #pragma unroll
        for (int t = 0; t < 16; ++t) {
            const v2f bf = *(const v2f*)(bp + 4 * t);   // b64, bank-conflict-free
            acc = __builtin_amdgcn_wmma_f32_16x16x4_f32(
                      false, a[t], false, bf, (short)0, acc, false, false);
        }
        const float en = snorm[c0 + col];
        const int   ci = c0 + col;
        #pragma unroll
        for (int i = 0; i < 8; ++i) {
            const float d = en - 2.0f * acc[i];
            if (d < best[i]) { best[i] = d; bidx[i] = ci; }
        }
    }

    // Reduce argmin across the 16 lanes sharing each C row (tie -> lower idx)
    #pragma unroll
    for (int i = 0; i < 8; ++i) {
        float bv = best[i];
        int   bi = bidx[i];
        #pragma unroll
        for (int mask = 1; mask < 16; mask <<= 1) {
            const float ov = __shfl_xor(bv, mask, 32);
            const int   oi = __shfl_xor(bi, mask, 32);
            if (ov < bv || (ov == bv && oi < bi)) { bv = ov; bi = oi; }
        }
        if (col == 0) {
            const int n = n0 + i + 8 * kg;   // C layout: M = i + 8*(lane>>4)
            wsIdx[n]   = bi;
            outIdxF[n] = (float)bi;
        }
    }
}

// ---------------------------------------------------------------------------
// Kernel B: gather emb[idx] into (B,L,H,W) layout + loss partial sums.
// float4 over contiguous w; l = (e>>12)&63; n = b*4096 + h*64 + w.
// ---------------------------------------------------------------------------
__global__ __launch_bounds__(256) void vq_gather_kernel(
    const float* __restrict__ ze, const float* __restrict__ emb,
    const int* __restrict__ wsIdx, float* __restrict__ outZ,
    float* __restrict__ wsLoss)
{
    const int t  = blockIdx.x * 256 + threadIdx.x;
    const int e0 = t << 2;
    const int w = e0 & 63, h = (e0 >> 6) & 63, l = (e0 >> 12) & 63, b = e0 >> 18;
    const int nb = (b << 12) + (h << 6) + w;

    const float4 z = *(const float4*)(ze + e0);
    float4 q;
    q.x = emb[(size_t)wsIdx[nb + 0] * 64 + l];
    q.y = emb[(size_t)wsIdx[nb + 1] * 64 + l];
    q.z = emb[(size_t)wsIdx[nb + 2] * 64 + l];
    q.w = emb[(size_t)wsIdx[nb + 3] * 64 + l];
    *(float4*)(outZ + e0) = q;   // z_q_st == z_q numerically

    const float dx = q.x - z.x, dy = q.y - z.y, dz = q.z - z.z, dw = q.w - z.w;
    float s = dx * dx + dy * dy + dz * dz + dw * dw;

    #pragma unroll
    for (int mask = 1; mask < 32; mask <<= 1) s += __shfl_xor(s, mask, 32);

    __shared__ float ps[8];
    if ((threadIdx.x & 31) == 0) ps[threadIdx.x >> 5] = s;
    __syncthreads();
    if (threadIdx.x == 0) {
        float tot = 0.f;
        #pragma unroll
        for (int i = 0; i < 8; ++i) tot += ps[i];
        atomicAdd(wsLoss, tot);
    }
}

__global__ void vq_init_kernel(float* wsLoss) {
    if (threadIdx.x == 0 && blockIdx.x == 0) *wsLoss = 0.f;
}

// loss = (1 + BETA) * sum((zq-ze)^2) / NELEM,  BETA = 0.25
__global__ void vq_final_kernel(const float* __restrict__ wsLoss,
                                float* __restrict__ outLoss) {
    *outLoss = *wsLoss * (1.25f / 4194304.0f);
}

// ---------------------------------------------------------------------------
extern "C" void kernel_launch(void* const* d_in, const int* in_sizes, int n_in,
                              void* d_out, int out_size, void* d_ws, size_t ws_size,
                              hipStream_t stream) {
    (void)in_sizes; (void)n_in; (void)out_size; (void)ws_size;

    const float* ze  = (const float*)d_in[0];   // (16,64,64,64) f32
    const float* emb = (const float*)d_in[1];   // (1024,64)     f32

    float* out      = (float*)d_out;
    float* outZ     = out;                 // [0, NELEM)
    float* outLoss  = out + NELEM;         // [NELEM]
    float* outIdxF  = out + NELEM + 1;     // [NELEM+1, NELEM+1+NVEC)

    float* wsLoss = (float*)d_ws;
    int*   wsIdx  = (int*)((char*)d_ws + 64);   // 64B-aligned scratch indices

    vq_init_kernel<<<1, 32, 0, stream>>>(wsLoss);

    const size_t smem = (size_t)(KCODES * PITCH + KCODES) * sizeof(float); // ~276 KB
    vq_argmin_kernel<<<NVEC / 128, 256, smem, stream>>>(ze, emb, wsIdx, outIdxF);

    vq_gather_kernel<<<NELEM / 1024, 256, 0, stream>>>(ze, emb, wsIdx, outZ, wsLoss);

    vq_final_kernel<<<1, 1, 0, stream>>>(wsLoss, outLoss);
}